// MyModel_87522843561288
// MI455X (gfx1250) — compile-verified
//
#include <hip/hip_runtime.h>
#include <math.h>

#define CROSS_EMB 16

typedef __attribute__((ext_vector_type(2))) float v2f;
typedef __attribute__((ext_vector_type(8))) float v8f;

__device__ __forceinline__ int lower_bound_i32(const int* __restrict__ a, int n, int v) {
    int lo = 0, hi = n;
    while (lo < hi) {
        int mid = (lo + hi) >> 1;
        if (a[mid] < v) lo = mid + 1; else hi = mid;
    }
    return lo;
}

// Full-wave32 sum via LDS-swizzle XOR butterflies (no LDS memory touched).
// group-of-32 encoding: offset = (xor_mask << 10) | (and_mask = 0x1F)
__device__ __forceinline__ float wave32_sum(float v) {
    v += __int_as_float(__builtin_amdgcn_ds_swizzle(__float_as_int(v), (1  << 10) | 0x1F));
    v += __int_as_float(__builtin_amdgcn_ds_swizzle(__float_as_int(v), (2  << 10) | 0x1F));
    v += __int_as_float(__builtin_amdgcn_ds_swizzle(__float_as_int(v), (4  << 10) | 0x1F));
    v += __int_as_float(__builtin_amdgcn_ds_swizzle(__float_as_int(v), (8  << 10) | 0x1F));
    v += __int_as_float(__builtin_amdgcn_ds_swizzle(__float_as_int(v), (16 << 10) | 0x1F));
    return v;
}

// One wave32 per segment (segment_ids sorted -> contiguous run found by a
// scalar binary search). Segment reduction on the tensor path:
// C(16x16) += A(16x4) x B(4x16), A row0 = ones, so C[0][n] accumulates the
// per-dim sums; a second accumulator takes B*B for the sum-of-squares.
// 8 rows per iteration = two B tiles -> 4 independent WMMA chains; ids loads
// are software-pipelined one iteration ahead to shorten the gather chain.
__global__ __launch_bounds__(256) void fm_segment_wmma_kernel(
    const int*   __restrict__ ids,          // [nnz]
    const int*   __restrict__ seg,          // [nnz], sorted
    const float* __restrict__ linear_bias,  // [1]
    const float* __restrict__ linear_w,     // [FEATURE_SIZE, 1]
    const float* __restrict__ cross_w,      // [FEATURE_SIZE, 16]
    float*       __restrict__ out,          // [3*batch]: logits | adjusted | prob
    int nnz, int batch)
{
    const int lane = (int)(threadIdx.x & 31);
    const int s = __builtin_amdgcn_readfirstlane(
                      (int)((blockIdx.x * blockDim.x + threadIdx.x) >> 5));
    if (s >= batch) return;                  // wave-uniform exit

    // Scalar binary search (s uniform -> s_load path) for [start, end).
    const int start = lower_bound_i32(seg, nnz, s);
    const int end   = lower_bound_i32(seg, nnz, s + 1);

    // A(16x4): lanes 0-15 M=lane {v0:K=0, v1:K=1}; lanes 16-31 M=lane-16
    // {v0:K=2, v1:K=3}. Row M=0 all ones -> lanes 0 and 16 carry 1.0.
    const float one_row = (lane == 0 || lane == 16) ? 1.0f : 0.0f;
    v2f a; a.x = one_row; a.y = one_row;

    v8f cs0 = {0.f,0.f,0.f,0.f,0.f,0.f,0.f,0.f};
    v8f cq0 = {0.f,0.f,0.f,0.f,0.f,0.f,0.f,0.f};
    v8f cs1 = {0.f,0.f,0.f,0.f,0.f,0.f,0.f,0.f};
    v8f cq1 = {0.f,0.f,0.f,0.f,0.f,0.f,0.f,0.f};

    const int   h  = lane >> 4;              // K half: {0,1} vs {2,3}
    const int   n  = lane & 15;              // embedding dim
    const float lane_lin_mask = (lane < 8) ? 1.0f : 0.0f;
    float lin = 0.0f;

    if (start < end) {
        const int e1 = end - 1;
        // Preload first tile's ids (clamped -> EXEC stays full, no OOB).
        int r0 = start + 2 * h;
        int i0 = ids[(r0     < end) ? r0     : e1];
        int i1 = ids[(r0 + 1 < end) ? r0 + 1 : e1];
        int i2 = ids[(r0 + 4 < end) ? r0 + 4 : e1];
        int i3 = ids[(r0 + 5 < end) ? r0 + 5 : e1];
        int rl = start + (lane & 7);
        int il = ids[(rl < end) ? rl : e1];

        for (int j = start; j < end; j += 8) {
            // Current-tile gathers (ids already resident from last iteration).
            float v0 = cross_w[i0 * CROSS_EMB + n];
            float v1 = cross_w[i1 * CROSS_EMB + n];
            float v2 = cross_w[i2 * CROSS_EMB + n];
            float v3 = cross_w[i3 * CROSS_EMB + n];
            float lv = linear_w[il];

            // Pipeline: issue next tile's ids loads before any WMMA waits.
            const int jn = j + 8;
            int p0 = jn + 2 * h;
            int N0 = ids[(p0     < end) ? p0     : e1];
            int N1 = ids[(p0 + 1 < end) ? p0 + 1 : e1];
            int N2 = ids[(p0 + 4 < end) ? p0 + 4 : e1];
            int N3 = ids[(p0 + 5 < end) ? p0 + 5 : e1];
            int pl = jn + (lane & 7);
            int Nl = ids[(pl < end) ? pl : e1];

            // Sequential prefetch of the upcoming ids cacheline (gfx1250
            // global_prefetch_b8) -- address needs no dependent load.
            {
                int pf = jn + 16 + (lane & 7);
                __builtin_prefetch(&ids[(pf < end) ? pf : e1], 0, 1);
            }

            // Tail masks for the current 8 rows.
            const int r0c = j + 2 * h;
            v0 = (r0c     < end) ? v0 : 0.0f;
            v1 = (r0c + 1 < end) ? v1 : 0.0f;
            v2 = (r0c + 4 < end) ? v2 : 0.0f;
            v3 = (r0c + 5 < end) ? v3 : 0.0f;

            v2f b0; b0.x = v0;      b0.y = v1;
            v2f b1; b1.x = v2;      b1.y = v3;
            v2f q0; q0.x = v0 * v0; q0.y = v1 * v1;
            v2f q1; q1.x = v2 * v2; q1.y = v3 * v3;

            // (neg_a, A, neg_b, B, c_mod, C, reuse_a, reuse_b)
            cs0 = __builtin_amdgcn_wmma_f32_16x16x4_f32(
                      false, a, false, b0, (short)0, cs0, false, false);
            cq0 = __builtin_amdgcn_wmma_f32_16x16x4_f32(
                      false, a, false, q0, (short)0, cq0, false, false);
            cs1 = __builtin_amdgcn_wmma_f32_16x16x4_f32(
                      false, a, false, b1, (short)0, cs1, false, false);
            cq1 = __builtin_amdgcn_wmma_f32_16x16x4_f32(
                      false, a, false, q1, (short)0, cq1, false, false);

            // Linear term, branchless: fmaf with a 0/1 mask keeps the gather
            // straight-line (no saveexec sinking).
            const int   rlc   = j + (lane & 7);
            const float lmask = ((rlc < end) ? 1.0f : 0.0f) * lane_lin_mask;
            lin = fmaf(lv, lmask, lin);

            i0 = N0; i1 = N1; i2 = N2; i3 = N3; il = Nl;
        }
    }

    // C layout: v0, lanes 0-15 -> (M=0, N=lane); rows M>0 are exactly zero
    // (A zero there), so lanes 16-31 contribute 0 below.
    const float sum_d = cs0[0] + cs1[0];
    const float sq_d  = cq0[0] + cq1[0];
    // Fold both reductions into one value: lanes 0-15 carry the FM term,
    // lanes 0-7 additionally carry linear partials, everything else is 0.
    float u = 0.5f * (sum_d * sum_d - sq_d) + lin;
    u = wave32_sum(u);

    if (lane == 0) {
        const float logit = linear_bias[0] + u;
        const float adj   = logit + logf(1.0f);     // NEG_SAMPLING_RATE = 1.0
        const float prob  = 1.0f / (1.0f + expf(-adj));
        out[s]             = logit;
        out[batch + s]     = adj;
        out[2 * batch + s] = prob;
    }
}

extern "C" void kernel_launch(void* const* d_in, const int* in_sizes, int n_in,
                              void* d_out, int out_size, void* d_ws, size_t ws_size,
                              hipStream_t stream) {
    // setup_inputs order: ids, segment_ids, batch_size, linear_bias, linear_w, cross_emb_w
    const int*   ids  = (const int*)d_in[0];
    const int*   seg  = (const int*)d_in[1];
    const float* bias = (const float*)d_in[3];
    const float* lw   = (const float*)d_in[4];
    const float* cw   = (const float*)d_in[5];
    float*       out  = (float*)d_out;

    const int nnz   = in_sizes[0];
    const int batch = out_size / 3;           // 3 outputs of [batch,1]

    const int threads = 256;                  // 8 wave32 per block
    const int wavesPerBlock = threads / 32;
    const int blocks = (batch + wavesPerBlock - 1) / wavesPerBlock;

    fm_segment_wmma_kernel<<<blocks, threads, 0, stream>>>(
        ids, seg, bias, lw, cw, out, nnz, batch);
}